// PQC6QDummy_74345883893945
// MI455X (gfx1250) — compile-verified
//
#include <hip/hip_runtime.h>
#include <math.h>

// Problem constants (from the reference)
#define N_QUBITS 6
#define BSZ      262144          // 2^18 batch rows
#define STATE_N  64              // 2^6 amplitudes
// Output = BSZ x STATE_N complex64 -> BSZ*STATE_N*2 floats = 128 MB (store-bound)

// Fixed reduce-kernel launch geometry -> compile-time trip count (enables unroll)
#define REDUCE_BLOCKS   256
#define REDUCE_THREADS  256
#define REDUCE_NWAVES   (REDUCE_BLOCKS * REDUCE_THREADS / 32)   // 2048
#define ROWS_PER_WAVE   (BSZ / REDUCE_NWAVES)                   // 128

typedef __attribute__((ext_vector_type(2))) float v2f;
typedef __attribute__((ext_vector_type(4))) float v4f;
typedef __attribute__((ext_vector_type(8))) float v8f;

// ---------------------------------------------------------------------------
// 0) zero the 6-float angle accumulator in workspace
// ---------------------------------------------------------------------------
__global__ void pqc_init_ws(float* __restrict__ accum) {
    if (threadIdx.x < 32) accum[threadIdx.x] = 0.0f;
}

// ---------------------------------------------------------------------------
// 1) theta[j] = sum_b x[b][j]  via chained V_WMMA_F32_16X16X4_F32.
//    A = ones(16x4)  ->  D[m][n] = sum_k B[k][n] + C[m][n]
//    Loads are UNCONDITIONAL from a clamped in-bounds address; invalid lanes
//    (n >= 6) are zeroed by a post-load multiply, so EXEC stays all-ones in
//    the loop. Compile-time trip count (ROWS_PER_WAVE) lets the compiler
//    unroll and keep multiple load clauses in flight per WMMA chain.
//    Exact in f32; independent of the B-operand K-slot permutation since A==1.
// ---------------------------------------------------------------------------
__global__ __launch_bounds__(REDUCE_THREADS)
void pqc_reduce_wmma(const float* __restrict__ x,
                     float* __restrict__ accum) {
    const unsigned lane  = threadIdx.x & 31u;
    const unsigned wave  = (blockIdx.x * REDUCE_THREADS + threadIdx.x) >> 5;
    const unsigned row0  = wave * ROWS_PER_WAVE;

    const unsigned n     = lane & 15u;                    // column (N) index
    const unsigned khalf = lane >> 4;                     // 0: K={0,1}, 1: K={2,3}
    const bool     valid = (n < N_QUBITS);
    const unsigned ncl   = valid ? n : 0u;                // clamped, always in-bounds
    const float    vmask = valid ? 1.0f : 0.0f;           // kills invalid lanes post-load

    const float* __restrict__ base = x + ncl;             // column base pointer
    const unsigned kA = 2u * khalf + 0u;
    const unsigned kB = 2u * khalf + 1u;

    v2f a; a.x = 1.0f; a.y = 1.0f;                        // all-ones A matrix
    v8f c = {};                                           // f32 accumulator (8 VGPRs)

    #pragma unroll 4
    for (unsigned i = 0; i < ROWS_PER_WAVE; i += 4) {     // 32 iterations, known at compile time
        const unsigned r = row0 + i;
        const float lx = base[(size_t)(r + kA) * N_QUBITS];
        const float ly = base[(size_t)(r + kB) * N_QUBITS];
        v2f b;
        b.x = vmask * lx;
        b.y = vmask * ly;
        // 8 args: (neg_a, A, neg_b, B, c_mod, C, reuse_a, reuse_b)
        c = __builtin_amdgcn_wmma_f32_16x16x4_f32(
                false, a, false, b, (short)0, c, false, false);
    }

    // D VGPR0: lanes 0-15 hold M=0, N=lane -> column sums for this wave's rows
    if (lane < N_QUBITS) atomicAdd(&accum[lane], c[0]);
}

// ---------------------------------------------------------------------------
// 2) state[k] = prod_i ( bit_i(k) ? -i*sin(theta_i/2) : cos(theta_i/2) )
//    wire 0 is the MSB.  Product of reals / pure-imaginaries:
//    state[k] = mag * (-i)^popcount(k).
// ---------------------------------------------------------------------------
__global__ void pqc_state(const float* __restrict__ accum,
                          float* __restrict__ state /* 128 floats, interleaved */) {
    const int k = threadIdx.x;                            // 0..63
    if (k >= STATE_N) return;
    float mag = 1.0f;
    int   pc  = 0;
    #pragma unroll
    for (int i = 0; i < N_QUBITS; ++i) {
        const int   bit  = (k >> (N_QUBITS - 1 - i)) & 1;
        const float half = 0.5f * accum[i];
        mag *= bit ? sinf(half) : cosf(half);
        pc  += bit;
    }
    float re, im;
    switch (pc & 3) {
        case 0:  re =  mag; im =  0.0f; break;
        case 1:  re =  0.0f; im = -mag; break;
        case 2:  re = -mag; im =  0.0f; break;
        default: re =  0.0f; im =  mag; break;
    }
    state[2 * k + 0] = re;
    state[2 * k + 1] = im;
}

// ---------------------------------------------------------------------------
// 3) Broadcast the 512-byte state to all BSZ rows: 128 MB of non-temporal
//    b128 streaming stores. Stride is a multiple of 32, so each thread's
//    source float4 (from LDS) is loop-invariant.
// ---------------------------------------------------------------------------
__global__ void pqc_broadcast(const v4f* __restrict__ state4,   // 32 x float4
                              v4f* __restrict__ out,
                              unsigned total4) {
    __shared__ v4f s[32];
    if (threadIdx.x < 32) s[threadIdx.x] = state4[threadIdx.x];
    __syncthreads();

    const unsigned idx    = blockIdx.x * blockDim.x + threadIdx.x;
    const unsigned stride = gridDim.x * blockDim.x;       // multiple of 32
    const v4f v = s[idx & 31u];

    for (unsigned j = idx; j < total4; j += stride) {
        __builtin_nontemporal_store(v, &out[j]);          // TH=NT global_store_b128
    }
}

// ---------------------------------------------------------------------------
extern "C" void kernel_launch(void* const* d_in, const int* in_sizes, int n_in,
                              void* d_out, int out_size, void* d_ws, size_t ws_size,
                              hipStream_t stream) {
    (void)in_sizes; (void)n_in; (void)out_size; (void)ws_size;

    const float* x     = (const float*)d_in[0];           // [BSZ, 6] f32
    float*       out   = (float*)d_out;                   // BSZ*64 complex64 (interleaved)
    float*       accum = (float*)d_ws;                    // 6 floats (padded)
    float*       state = accum + 32;                      // 128 floats, 16B aligned

    pqc_init_ws<<<1, 64, 0, stream>>>(accum);

    // 256 blocks x 256 threads = 2048 waves -> 128 rows/wave, 32 WMMA/wave
    pqc_reduce_wmma<<<REDUCE_BLOCKS, REDUCE_THREADS, 0, stream>>>(x, accum);

    pqc_state<<<1, 64, 0, stream>>>(accum, state);

    const unsigned total4 = (unsigned)BSZ * STATE_N * 2u / 4u;   // 8,388,608 float4s
    pqc_broadcast<<<4096, 256, 0, stream>>>((const v4f*)state, (v4f*)out, total4);
}